// Model_32933809226431
// MI455X (gfx1250) — compile-verified
//
#include <hip/hip_runtime.h>
#include <math.h>

// ---------------------------------------------------------------------------
// CDNA5 (gfx1250) implementation. Convolutions run on the fp32 matrix path
// (V_WMMA_F32_16X16X4_F32) via implicit GEMM, wave32.  Each wave computes a
// 16(cout) x 64(pixel) tile: one A (weight) fragment feeds 4 WMMAs.
// ---------------------------------------------------------------------------

typedef __attribute__((ext_vector_type(2))) float v2f;
typedef __attribute__((ext_vector_type(8))) float v8f;

struct BNP { const float *beta, *gamma, *mean, *var; };

__device__ __forceinline__ float selu_f(float x) {
    const float sc = 1.0507009873554805f, al = 1.6732632423543772f;
    return x > 0.f ? sc * x : sc * al * (expf(x) - 1.f);
}
__device__ __forceinline__ float bn_f(float x, const BNP& p, int c) {
    return p.gamma[c] * (x - p.mean[c]) * rsqrtf(p.var[c] + 1e-5f) + p.beta[c];
}

// ---------------------------------------------------------------------------
// Elementwise BatchNorm (+ optional SELU).  in/out may alias.
// ---------------------------------------------------------------------------
__global__ void bn_act_kernel(const float* __restrict__ in, float* __restrict__ out,
                              const float* __restrict__ beta, const float* __restrict__ gamma,
                              const float* __restrict__ mean, const float* __restrict__ var,
                              int C, int HW, int do_selu, long long total) {
    long long i = (long long)blockIdx.x * blockDim.x + threadIdx.x;
    if (i >= total) return;
    int c = (int)((i / HW) % C);
    float v = gamma[c] * (in[i] - mean[c]) * rsqrtf(var[c] + 1e-5f) + beta[c];
    out[i] = do_selu ? selu_f(v) : v;
}

__global__ void add_kernel(float* __restrict__ y, const float* __restrict__ x, long long n) {
    long long i = (long long)blockIdx.x * blockDim.x + threadIdx.x;
    if (i < n) y[i] += x[i];
}

// 2x2 stride-2 max pool over (BC, H, W)
__global__ void maxpool_kernel(const float* __restrict__ in, float* __restrict__ out,
                               int H, int W, long long totalOut) {
    long long i = (long long)blockIdx.x * blockDim.x + threadIdx.x;
    if (i >= totalOut) return;
    int OW = W >> 1, OH = H >> 1;
    int ox = (int)(i % OW);
    long long r = i / OW;
    int oy = (int)(r % OH);
    long long bc = r / OH;
    const float* p = in + (bc * H + 2 * oy) * (long long)W + 2 * ox;
    out[i] = fmaxf(fmaxf(p[0], p[1]), fmaxf(p[W], p[W + 1]));
}

// ---------------------------------------------------------------------------
// Implicit-GEMM convolution on V_WMMA_F32_16X16X4_F32.
//   in : (B, Cin, H, W)   w : (Cout, Cin, KH, KW) (= w[m*Ktot + k])  bias:(Cout)
//   out: (B, Cout, H, W)  pad = K/2 ("same"),  W = 1<<lw (power of two)
// One wave per 16-cout x 64-pixel tile (pixels = flattened y*W+x; H*W is a
// multiple of 64 for every layer).  Lane mapping (ISA 7.12.2):
//   A 16x4 : lane&15 -> M row, lane half selects K{0,1}/{2,3} in VGPR0/1
//   B 4x16 : lane&15 -> N col, lane half selects K{0,1}/{2,3}
//   C/D    : VGPR r, lane<16 -> (M=r, N=lane); lane>=16 -> (M=r+8, N=lane-16)
// EXEC is all-ones at each WMMA: blockDim==32, no early returns, exact grid.
// ---------------------------------------------------------------------------
template <int KH, int KW>
__global__ void conv_wmma_kernel(const float* __restrict__ in, const float* __restrict__ w,
                                 const float* __restrict__ bias, float* __restrict__ out,
                                 int Cin, int Cout, int H, int lw) {
    const int W  = 1 << lw;
    const int PH = KH / 2, PW = KW / 2;
    const int lane  = threadIdx.x;
    const int n15   = lane & 15;
    const int half  = lane >> 4;
    const int pbase = blockIdx.x << 6;          // first of 64 pixels
    const int coB   = blockIdx.y << 4;
    const int b     = blockIdx.z;
    const int Ktot  = Cin * KH * KW;
    const size_t HW = (size_t)H << lw;
    const float* inb = in + (size_t)b * Cin * HW;
    const int mrow = coB + n15;                 // A row (output channel)

    v8f acc[4];
#pragma unroll
    for (int t = 0; t < 4; ++t)
#pragma unroll
        for (int r = 0; r < 8; ++r) acc[t][r] = 0.f;

    for (int k0 = 0; k0 < Ktot; k0 += 4) {
        int ka = k0 + (half << 1);
        v2f A;
        A.x = (ka     < Ktot) ? w[(size_t)mrow * Ktot + ka]     : 0.f;
        A.y = (ka + 1 < Ktot) ? w[(size_t)mrow * Ktot + ka + 1] : 0.f;
        v2f Bv[4];
#pragma unroll
        for (int j = 0; j < 2; ++j) {
            int k = ka + j;
            bool kv = (k < Ktot);
            int ci = 0, kh = 0, kw_ = 0;
            if (kv) {
                ci = k / (KH * KW);
                int rem = k - ci * (KH * KW);
                kh  = rem / KW;
                kw_ = rem - kh * KW;
            }
            const float* src = inb + (size_t)ci * HW;
#pragma unroll
            for (int t = 0; t < 4; ++t) {
                int p  = pbase + (t << 4) + n15;
                int y  = p >> lw, x = p & (W - 1);
                int iy = y + kh - PH, ix = x + kw_ - PW;
                float v = 0.f;
                if (kv && iy >= 0 && iy < H && ix >= 0 && ix < W)
                    v = src[((size_t)iy << lw) + ix];
                if (j == 0) Bv[t].x = v; else Bv[t].y = v;
            }
        }
#pragma unroll
        for (int t = 0; t < 4; ++t)
            acc[t] = __builtin_amdgcn_wmma_f32_16x16x4_f32(false, A, false, Bv[t],
                                                           (short)0, acc[t], false, false);
    }

    float bv[8];
#pragma unroll
    for (int r = 0; r < 8; ++r) bv[r] = bias[coB + r + (half << 3)];
    float* ob = out + (size_t)b * Cout * HW;
#pragma unroll
    for (int t = 0; t < 4; ++t) {
        int p = pbase + (t << 4) + n15;
#pragma unroll
        for (int r = 0; r < 8; ++r) {
            int m = coB + r + (half << 3);
            ob[(size_t)m * HW + p] = acc[t][r] + bv[r];
        }
    }
}

// ---------------------------------------------------------------------------
// e_T[b,t,c] = max_f |h[b,c,f,t]| ; e_S[b,f,c] = max_t |h[b,c,f,t]|
// h : (16, 64, 2, 8).  grid=16, block=64 (thread = channel c)
// ---------------------------------------------------------------------------
__global__ void feat_reduce_kernel(const float* __restrict__ h,
                                   float* __restrict__ eT, float* __restrict__ eS) {
    int b = blockIdx.x, c = threadIdx.x;
    const float* hb = h + ((size_t)b * 64 + c) * 16;
    for (int f = 0; f < 2; ++f) {
        float mt = 0.f;
        for (int t = 0; t < 8; ++t) mt = fmaxf(mt, fabsf(hb[f * 8 + t]));
        eS[((size_t)b * 2 + f) * 64 + c] = mt;
    }
    for (int t = 0; t < 8; ++t)
        eT[((size_t)b * 8 + t) * 64 + c] = fmaxf(fabsf(hb[t]), fabsf(hb[8 + t]));
}

// ---------------------------------------------------------------------------
// GAT pooling head.  x:(bs,N,64) -> out:(bs,64).  grid=bs, block=64 (=dout).
// Cross-lane attention-score reduction via LDS ds_add_f32 atomics.
// ---------------------------------------------------------------------------
struct GatArgs {
    const float* x; int N; int din; int dout;
    const float *att_w, *att_b, *att_v, *pw_w, *pw_b, *po_w, *po_b;
    BNP bn; float temp; float* out;
};
__global__ void gat_kernel(GatArgs A) {
    int b = blockIdx.x, d = threadIdx.x;
    __shared__ float xs[8 * 64];
    __shared__ float sc[8];
    const float* xb = A.x + (size_t)b * A.N * A.din;
    for (int i = d; i < A.N * A.din; i += blockDim.x) xs[i] = xb[i];
    if (d < A.N) sc[d] = 0.f;
    __syncthreads();
    for (int n = 0; n < A.N; ++n) {
        float u = A.att_b[d];
        for (int k = 0; k < A.din; ++k) u += xs[n * A.din + k] * A.att_w[k * A.dout + d];
        atomicAdd(&sc[n], tanhf(u) * A.att_v[d]);
    }
    __syncthreads();
    float mx = -1e30f;
    for (int n = 0; n < A.N; ++n) mx = fmaxf(mx, sc[n] / A.temp);
    float den = 0.f, a[8];
    for (int n = 0; n < A.N; ++n) { a[n] = expf(sc[n] / A.temp - mx); den += a[n]; }
    float p = 0.f;
    for (int n = 0; n < A.N; ++n) {
        float u = A.pw_b[d];
        for (int k = 0; k < A.din; ++k) u += xs[n * A.din + k] * A.pw_w[k * A.dout + d];
        p += u * (a[n] / den);
    }
    float q = A.po_b[d];
    for (int k = 0; k < A.din; ++k) q += xs[k] * A.po_w[k * A.dout + d];
    A.out[(size_t)b * A.dout + d] = selu_f(bn_f(p + q, A.bn, d));
}

// ---------------------------------------------------------------------------
// HTRG block.  Three attention-pooled projections share proj_w; outputs are
// selu(bn_X(pT+pS+pM)).  grid=bs, block=32 (=dout).
// ---------------------------------------------------------------------------
struct HtrgArgs {
    const float *xT, *xS, *xm; int NT, NS, xmStride, din;
    const float *apT_w, *apT_b, *avT, *apS_w, *apS_b, *avS, *apM_w, *apM_b, *avM;
    const float *proj_w, *proj_b;
    BNP bnT, bnS, bnM;
    float *outT, *outS, *outM; float temp;
};
__device__ float att_pool(const float* xs, int N, int din, int dout, int d,
                          const float* apw, const float* apb, const float* av,
                          const float* pjw, const float* pjb, float* sc, float temp) {
    if (d < N) sc[d] = 0.f;
    __syncthreads();
    for (int n = 0; n < N; ++n) {
        float u = apb[d];
        for (int k = 0; k < din; ++k) u += xs[n * din + k] * apw[k * dout + d];
        atomicAdd(&sc[n], tanhf(u) * av[d]);
    }
    __syncthreads();
    float mx = -1e30f;
    for (int n = 0; n < N; ++n) mx = fmaxf(mx, sc[n] / temp);
    float den = 0.f, a[2];
    for (int n = 0; n < N; ++n) { a[n] = expf(sc[n] / temp - mx); den += a[n]; }
    float p = 0.f;
    for (int n = 0; n < N; ++n) {
        float u = pjb[d];
        for (int k = 0; k < din; ++k) u += xs[n * din + k] * pjw[k * dout + d];
        p += u * (a[n] / den);
    }
    return p;
}
__global__ void htrg_kernel(HtrgArgs A) {
    int b = blockIdx.x, d = threadIdx.x;
    const int dout = 32;
    __shared__ float xT[2 * 64], xS[2 * 64], xM[64];
    __shared__ float scT[2], scS[2], scM[2];
    const float* pT = A.xT + (size_t)b * A.NT * A.din;
    const float* pS = A.xS + (size_t)b * A.NS * A.din;
    const float* pM = A.xm + (size_t)b * A.xmStride;
    for (int i = d; i < A.NT * A.din; i += dout) xT[i] = pT[i];
    for (int i = d; i < A.NS * A.din; i += dout) xS[i] = pS[i];
    for (int i = d; i < A.din;        i += dout) xM[i] = pM[i];
    __syncthreads();
    float vT = att_pool(xT, A.NT, A.din, dout, d, A.apT_w, A.apT_b, A.avT, A.proj_w, A.proj_b, scT, A.temp);
    float vS = att_pool(xS, A.NS, A.din, dout, d, A.apS_w, A.apS_b, A.avS, A.proj_w, A.proj_b, scS, A.temp);
    float vM = att_pool(xM, 1,    A.din, dout, d, A.apM_w, A.apM_b, A.avM, A.proj_w, A.proj_b, scM, A.temp);
    float agg = vT + vS + vM;
    A.outT[(size_t)b * dout + d] = selu_f(bn_f(agg, A.bnT, d));
    A.outS[(size_t)b * dout + d] = selu_f(bn_f(agg, A.bnS, d));
    A.outM[(size_t)b * dout + d] = selu_f(bn_f(agg, A.bnM, d));
}

// graph_pool(x, 0.5) on a single-node input == duplicate node. (bs,1,d)->(bs,2,d)
__global__ void dup2_kernel(const float* __restrict__ in, float* __restrict__ out, int d) {
    int b = blockIdx.x, t = threadIdx.x;
    float v = in[(size_t)b * d + t];
    out[((size_t)b * 2 + 0) * d + t] = v;
    out[((size_t)b * 2 + 1) * d + t] = v;
}
// y(bs,n,d) += a(bs,1,d)
__global__ void add_bcast_kernel(float* __restrict__ y, const float* __restrict__ a, int n, int d) {
    int b = blockIdx.x, t = threadIdx.x;
    float v = a[(size_t)b * d + t];
    for (int i = 0; i < n; ++i) y[((size_t)b * n + i) * d + t] += v;
}

// ---------------------------------------------------------------------------
// Final: elementwise max of branches, feature concat, 160x2 linear.
// d_out = feats(16,160) || logits(16,2).  grid=16, block=32.
// ---------------------------------------------------------------------------
__global__ void final_kernel(const float* t1, const float* t2, const float* s1, const float* s2,
                             const float* m1, const float* m2,
                             const float* __restrict__ ow, const float* __restrict__ ob,
                             float* __restrict__ dout) {
    int b = blockIdx.x, d = threadIdx.x;
    __shared__ float feats[160];
    const float* T1 = t1 + (size_t)b * 64; const float* T2 = t2 + (size_t)b * 64;
    const float* S1 = s1 + (size_t)b * 64; const float* S2 = s2 + (size_t)b * 64;
    float ta0 = fmaxf(T1[d], T2[d]),       ta1 = fmaxf(T1[32 + d], T2[32 + d]);
    float sa0 = fmaxf(S1[d], S2[d]),       sa1 = fmaxf(S1[32 + d], S2[32 + d]);
    feats[d]        = fmaxf(fabsf(ta0), fabsf(ta1));
    feats[32 + d]   = 0.5f * (ta0 + ta1);
    feats[64 + d]   = fmaxf(fabsf(sa0), fabsf(sa1));
    feats[96 + d]   = 0.5f * (sa0 + sa1);
    feats[128 + d]  = fmaxf(m1[(size_t)b * 32 + d], m2[(size_t)b * 32 + d]);
    __syncthreads();
    for (int j = d; j < 160; j += 32) dout[(size_t)b * 160 + j] = feats[j];
    if (d < 2) {
        float acc = ob[d];
        for (int k = 0; k < 160; ++k) acc += feats[k] * ow[k * 2 + d];
        dout[16 * 160 + (size_t)b * 2 + d] = acc;
    }
}

// ---------------------------------------------------------------------------
// Host orchestration.
// Input order assumption: top-level setup_inputs() insertion order (x, params),
// nested params flattened in JAX tree order (dict keys sorted at every level).
// ---------------------------------------------------------------------------
extern "C" void kernel_launch(void* const* d_in, const int* in_sizes, int n_in,
                              void* d_out, int out_size, void* d_ws, size_t ws_size,
                              hipStream_t stream) {
    (void)in_sizes; (void)n_in; (void)out_size; (void)ws_size;
    int idx = 0;
    auto IN = [&](void) { return (const float*)d_in[idx++]; };
    auto rdBN = [&](void) { BNP p; p.beta = IN(); p.gamma = IN(); p.mean = IN(); p.var = IN(); return p; };

    const float* x_in = IN();                       // x (16,128,512)

    // ---- params['blocks'] : list of 6 dicts (keys sorted per dict) ----
    struct ResP { BNP bn1; bool has_bn1; BNP bn2; const float *c1b, *c1w, *c2b, *c2w;
                  bool down; const float* db; BNP dbn; const float* dw; } rb[6];
    // blocks[0]: bn2, conv1_b, conv1_w, conv2_b, conv2_w
    rb[0].has_bn1 = false; rb[0].down = false;
    rb[0].bn2 = rdBN(); rb[0].c1b = IN(); rb[0].c1w = IN(); rb[0].c2b = IN(); rb[0].c2w = IN();
    // blocks[1]: bn1, bn2, conv1_b, conv1_w, conv2_b, conv2_w, down_b, down_bn, down_w
    rb[1].has_bn1 = true; rb[1].down = true;
    rb[1].bn1 = rdBN(); rb[1].bn2 = rdBN();
    rb[1].c1b = IN(); rb[1].c1w = IN(); rb[1].c2b = IN(); rb[1].c2w = IN();
    rb[1].db = IN(); rb[1].dbn = rdBN(); rb[1].dw = IN();
    // blocks[2..5]: bn1, bn2, conv1_b, conv1_w, conv2_b, conv2_w
    for (int i = 2; i < 6; ++i) {
        rb[i].has_bn1 = true; rb[i].down = false;
        rb[i].bn1 = rdBN(); rb[i].bn2 = rdBN();
        rb[i].c1b = IN(); rb[i].c1w = IN(); rb[i].c2b = IN(); rb[i].c2w = IN();
    }
    const float* ct_b = IN(); BNP ct_bn = rdBN(); const float* ct_w = IN(); BNP first_bn = rdBN();

    struct GatP { const float *att_b, *att_v, *att_w; BNP bn; const float *po_b, *po_w, *pw_b, *pw_w; };
    auto rdGat = [&](void) { GatP g; g.att_b = IN(); g.att_v = IN(); g.att_w = IN();
                             g.bn = rdBN(); g.po_b = IN(); g.po_w = IN(); g.pw_b = IN(); g.pw_w = IN(); return g; };
    GatP gatS = rdGat(), gatT = rdGat();
    const float* master1 = IN(); const float* master2 = IN();
    const float* out_b = IN();   const float* out_w = IN();

    struct HtrgP { const float *apM_b, *apM_w, *apS_b, *apS_w, *apT_b, *apT_w, *avM, *avS, *avT;
                   BNP bnM, bnS, bnT; const float *proj_b, *proj_w; };
    auto rdHtrg = [&](void) { HtrgP h;
        h.apM_b = IN(); h.apM_w = IN(); h.apS_b = IN(); h.apS_w = IN(); h.apT_b = IN(); h.apT_w = IN();
        h.avM = IN(); h.avS = IN(); h.avT = IN();
        h.bnM = rdBN(); h.bnS = rdBN(); h.bnT = rdBN();
        h.proj_b = IN(); h.proj_w = IN(); return h; };
    HtrgP st11 = rdHtrg(), st12 = rdHtrg(), st21 = rdHtrg(), st22 = rdHtrg();

    // ---- workspace layout: 3 rotating 134MB regions + small area ----
    const size_t BIG = (size_t)16 * 32 * 128 * 512;     // 33,554,432 floats
    float* W0 = (float*)d_ws;
    float* RA = W0; float* RB = W0 + BIG; float* RC = W0 + 2 * BIG;
    float* S = W0 + 3 * BIG;
    float* eT  = S; S += 16 * 8 * 64;
    float* eS  = S; S += 16 * 2 * 64;
    float* gTb = S; S += 16 * 64;
    float* gSb = S; S += 16 * 64;
    float* h1t = S; S += 512; float* h1s = S; S += 512; float* h1m = S; S += 512;
    float* t1d = S; S += 1024; float* s1d = S; S += 1024;
    float* ta1 = S; S += 512; float* sa1 = S; S += 512; float* ma1 = S; S += 512;
    float* h2t = S; S += 512; float* h2s = S; S += 512; float* h2m = S; S += 512;
    float* t2d = S; S += 1024; float* s2d = S; S += 1024;
    float* ta2 = S; S += 512; float* sa2 = S; S += 512; float* ma2 = S; S += 512;

    // ---- stem: first_bn -> conv5x5(1->32) -> bn -> selu ----
    {
        long long tot = 16LL * 128 * 512;
        bn_act_kernel<<<(unsigned)((tot + 255) / 256), 256, 0, stream>>>(
            x_in, RA, first_bn.beta, first_bn.gamma, first_bn.mean, first_bn.var,
            1, 128 * 512, 0, tot);
        conv_wmma_kernel<5, 5><<<dim3((128 * 512) / 64, 32 / 16, 16), 32, 0, stream>>>(
            RA, ct_w, ct_b, RB, 1, 32, 128, 9);
        long long tot1 = 16LL * 32 * 128 * 512;
        bn_act_kernel<<<(unsigned)((tot1 + 255) / 256), 256, 0, stream>>>(
            RB, RB, ct_bn.beta, ct_bn.gamma, ct_bn.mean, ct_bn.var,
            32, 128 * 512, 1, tot1);
    }

    // ---- residual blocks ----
    int cins[6]  = {32, 32, 64, 64, 64, 64};
    int couts[6] = {32, 64, 64, 64, 64, 64};
    int Hc = 128, Wc = 512, lw = 9;
    float* X = RB; float* Y = RA; float* Z = RC;   // X holds current h
    for (int i = 0; i < 6; ++i) {
        int ci = cins[i], co = couts[i];
        long long hw = (long long)Hc * Wc;
        unsigned ptiles = (unsigned)(hw / 64);
        const float* pre = X;
        if (rb[i].has_bn1) {
            long long t = 16LL * ci * hw;
            bn_act_kernel<<<(unsigned)((t + 255) / 256), 256, 0, stream>>>(
                X, Y, rb[i].bn1.beta, rb[i].bn1.gamma, rb[i].bn1.mean, rb[i].bn1.var,
                ci, (int)hw, 1, t);
            pre = Y;
        }
        conv_wmma_kernel<3, 3><<<dim3(ptiles, co / 16, 16), 32, 0, stream>>>(
            pre, rb[i].c1w, rb[i].c1b, Z, ci, co, Hc, lw);
        long long tc = 16LL * co * hw;
        bn_act_kernel<<<(unsigned)((tc + 255) / 256), 256, 0, stream>>>(
            Z, Z, rb[i].bn2.beta, rb[i].bn2.gamma, rb[i].bn2.mean, rb[i].bn2.var,
            co, (int)hw, 1, tc);
        conv_wmma_kernel<3, 3><<<dim3(ptiles, co / 16, 16), 32, 0, stream>>>(
            Z, rb[i].c2w, rb[i].c2b, Y, co, co, Hc, lw);
        if (rb[i].down) {
            conv_wmma_kernel<1, 1><<<dim3(ptiles, co / 16, 16), 32, 0, stream>>>(
                X, rb[i].dw, rb[i].db, Z, ci, co, Hc, lw);
            bn_act_kernel<<<(unsigned)((tc + 255) / 256), 256, 0, stream>>>(
                Z, Z, rb[i].dbn.beta, rb[i].dbn.gamma, rb[i].dbn.mean, rb[i].dbn.var,
                co, (int)hw, 0, tc);
            add_kernel<<<(unsigned)((tc + 255) / 256), 256, 0, stream>>>(Y, Z, tc);
        } else {
            add_kernel<<<(unsigned)((tc + 255) / 256), 256, 0, stream>>>(Y, X, tc);
        }
        long long to = 16LL * co * (Hc / 2) * (Wc / 2);
        maxpool_kernel<<<(unsigned)((to + 255) / 256), 256, 0, stream>>>(Y, Z, Hc, Wc, to);
        float* t = X; X = Z; Z = Y; Y = t;          // rotate; new h in X
        Hc /= 2; Wc /= 2; lw -= 1;
    }
    const float* hfin = X;                           // (16, 64, 2, 8)

    // ---- graph heads ----
    feat_reduce_kernel<<<16, 64, 0, stream>>>(hfin, eT, eS);

    GatArgs gaT; gaT.x = eT; gaT.N = 8; gaT.din = 64; gaT.dout = 64;
    gaT.att_w = gatT.att_w; gaT.att_b = gatT.att_b; gaT.att_v = gatT.att_v;
    gaT.pw_w = gatT.pw_w; gaT.pw_b = gatT.pw_b; gaT.po_w = gatT.po_w; gaT.po_b = gatT.po_b;
    gaT.bn = gatT.bn; gaT.temp = 2.0f; gaT.out = gTb;
    gat_kernel<<<16, 64, 0, stream>>>(gaT);

    GatArgs gaS = gaT; gaS.x = eS; gaS.N = 2;
    gaS.att_w = gatS.att_w; gaS.att_b = gatS.att_b; gaS.att_v = gatS.att_v;
    gaS.pw_w = gatS.pw_w; gaS.pw_b = gatS.pw_b; gaS.po_w = gatS.po_w; gaS.po_b = gatS.po_b;
    gaS.bn = gatS.bn; gaS.out = gSb;
    gat_kernel<<<16, 64, 0, stream>>>(gaS);

    auto fillH = [](HtrgArgs& a, const HtrgP& p) {
        a.apT_w = p.apT_w; a.apT_b = p.apT_b; a.avT = p.avT;
        a.apS_w = p.apS_w; a.apS_b = p.apS_b; a.avS = p.avS;
        a.apM_w = p.apM_w; a.apM_b = p.apM_b; a.avM = p.avM;
        a.proj_w = p.proj_w; a.proj_b = p.proj_b;
        a.bnT = p.bnT; a.bnS = p.bnS; a.bnM = p.bnM; a.temp = 2.0f;
    };
    auto run_branch = [&](const HtrgP& p1, const HtrgP& p2, const float* master,
                          float* ht, float* hs, float* hm, float* td, float* sd,
                          float* ta, float* sa, float* ma) {
        HtrgArgs a1; fillH(a1, p1);
        a1.xT = gTb; a1.NT = 1; a1.xS = gSb; a1.NS = 1;
        a1.xm = master; a1.xmStride = 0; a1.din = 64;
        a1.outT = ht; a1.outS = hs; a1.outM = hm;
        htrg_kernel<<<16, 32, 0, stream>>>(a1);
        dup2_kernel<<<16, 32, 0, stream>>>(ht, td, 32);
        dup2_kernel<<<16, 32, 0, stream>>>(hs, sd, 32);
        HtrgArgs a2; fillH(a2, p2);
        a2.xT = td; a2.NT = 2; a2.xS = sd; a2.NS = 2;
        a2.xm = hm; a2.xmStride = 32; a2.din = 32;
        a2.outT = ta; a2.outS = sa; a2.outM = ma;
        htrg_kernel<<<16, 32, 0, stream>>>(a2);
        add_bcast_kernel<<<16, 32, 0, stream>>>(td, ta, 2, 32);
        add_bcast_kernel<<<16, 32, 0, stream>>>(sd, sa, 2, 32);
        add_bcast_kernel<<<16, 32, 0, stream>>>(hm, ma, 1, 32);
    };
    run_branch(st11, st12, master1, h1t, h1s, h1m, t1d, s1d, ta1, sa1, ma1);
    run_branch(st21, st22, master2, h2t, h2s, h2m, t2d, s2d, ta2, sa2, ma2);

    final_kernel<<<16, 32, 0, stream>>>(t1d, t2d, s1d, s2d, h1m, h2m, out_w, out_b, (float*)d_out);
}